// RasterPoints_40724879901150
// MI455X (gfx1250) — compile-verified
//
#include <hip/hip_runtime.h>

// RasterPoints: out[b,t,row,col,p] = 1.0, zeros elsewhere.
// B=16, T=50, H=W=100, P=25  ->  out = 200,000,000 floats (800 MB).
// Pure store-bandwidth problem: floor ~34us at 23.3 TB/s. No matrix math
// exists in this op, so no WMMA; the optimal CDNA5 path is wave32 b128
// non-temporal stores (output is 4x the 192MB L2 -> stream, don't cache).

#define SDF_H 100
#define SDF_W 100
#define NPTS  25           // P
#define XSTRIDE (2 * NPTS) // floats of x per (b,t)
#define FILL_UNROLL 4      // b128 stores per thread

typedef float float4v __attribute__((ext_vector_type(4)));

// ---------------------------------------------------------------------------
// Pass 1: zero-fill with 128-bit non-temporal stores, 4 per thread.
// Grid-stride pattern: iteration k stores at (gid + k*stride), so consecutive
// lanes always hit consecutive 16B chunks -> every global_store_b128 issued
// by a wave is a fully coalesced 512B burst. No tail logic in-kernel.
// ---------------------------------------------------------------------------
__global__ void raster_fill_zero(float4v* __restrict__ out, long long nvec4) {
    const long long gid    = (long long)blockIdx.x * blockDim.x + threadIdx.x;
    const long long stride = (long long)gridDim.x * blockDim.x;
    const float4v z = {0.0f, 0.0f, 0.0f, 0.0f};
#pragma unroll
    for (int k = 0; k < FILL_UNROLL; ++k) {
        const long long i = gid + (long long)k * stride;
        if (i < nvec4) {
            __builtin_nontemporal_store(z, &out[i]);  // global_store_b128 th:NT
        }
    }
}

// Tiny scalar tail fill (only launched if out_size % 4 != 0; never for this
// shape, but kept for generality).
__global__ void raster_fill_tail(float* __restrict__ out, long long base, int count) {
    int i = blockIdx.x * blockDim.x + threadIdx.x;
    if (i < count) out[base + i] = 0.0f;
}

// ---------------------------------------------------------------------------
// Pass 2: scatter the 20,000 ones. One thread per (b,t,p) point.
//   col = (int)(x[b,t,2p]   / res[b,t,0] + org[b,t,0])
//   row = (int)(x[b,t,2p+1] / res[b,t,1] + org[b,t,1])
// Matches the reference's f32 divide + truncating int cast. OOB indices are
// dropped (JAX scatter default); data guarantees [25,75) so none occur.
// ---------------------------------------------------------------------------
__global__ void raster_scatter_ones(const float* __restrict__ x,
                                    const float* __restrict__ res,
                                    const float* __restrict__ org,
                                    float* __restrict__ out,
                                    int total) {
    int tid = blockIdx.x * blockDim.x + threadIdx.x;
    if (tid >= total) return;

    int p  = tid % NPTS;
    int bt = tid / NPTS;   // flattened b*T + t

    const float px = x[bt * XSTRIDE + 2 * p + 0];
    const float py = x[bt * XSTRIDE + 2 * p + 1];
    const float rx = res[bt * 2 + 0];
    const float ry = res[bt * 2 + 1];
    const float ox = org[bt * 2 + 0];
    const float oy = org[bt * 2 + 1];

    const int col = (int)(px / rx + ox);
    const int row = (int)(py / ry + oy);

    if (row < 0 || row >= SDF_H || col < 0 || col >= SDF_W) return; // drop OOB

    const long long o =
        ((((long long)bt * SDF_H + row) * SDF_W + col) * NPTS + p);
    out[o] = 1.0f;
}

// ---------------------------------------------------------------------------
extern "C" void kernel_launch(void* const* d_in, const int* in_sizes, int n_in,
                              void* d_out, int out_size, void* d_ws, size_t ws_size,
                              hipStream_t stream) {
    (void)n_in; (void)d_ws; (void)ws_size;

    const float* x   = (const float*)d_in[0];  // [B, T, 2P]
    const float* res = (const float*)d_in[1];  // [B, T, 2]
    const float* org = (const float*)d_in[2];  // [B, T, 2]
    float*       out = (float*)d_out;          // [B, T, H, W, P]

    const long long nelem = (long long)out_size;       // 200,000,000
    const long long nvec4 = nelem / 4;                 // 50,000,000
    const int  tail_count = (int)(nelem - nvec4 * 4);  // 0 for this shape

    // Pass 1: stream zeros to HBM (write-bandwidth bound, ~34us floor).
    {
        const int threads = 256;                        // 8 wave32 per block
        const long long per_block = (long long)threads * FILL_UNROLL;
        const long long blocks = (nvec4 + per_block - 1) / per_block;  // ~48.8K
        raster_fill_zero<<<dim3((unsigned)blocks), threads, 0, stream>>>(
            (float4v*)out, nvec4);
    }
    if (tail_count > 0) {
        raster_fill_tail<<<1, 64, 0, stream>>>(out, nvec4 * 4, tail_count);
    }

    // Pass 2: 20,000 scattered ones. In-order stream (and the captured graph's
    // dependency chain) guarantees this lands after the fill.
    {
        const int total = (in_sizes[0] / XSTRIDE) * NPTS;  // B*T*P = 20,000
        const int threads = 256;
        const int blocks = (total + threads - 1) / threads;
        raster_scatter_ones<<<blocks, threads, 0, stream>>>(x, res, org, out, total);
    }
}